// FrameLogLikelihood_28862180229353
// MI455X (gfx1250) — compile-verified
//
#include <hip/hip_runtime.h>
#include <hip/hip_bf16.h>

typedef __attribute__((ext_vector_type(2))) float v2f;
typedef __attribute__((ext_vector_type(4))) float v4f;
typedef __attribute__((ext_vector_type(8))) float v8f;

#define K_SEQ       2001
#define M_DIM       16
#define SEQ_ELEMS   (K_SEQ * M_DIM)      // 32016 floats per sequence
#define ROWS_TILE   64
#define TILE_ELEMS  (ROWS_TILE * M_DIM)  // 1024 floats = 4KB per tile
#define NUM_TILES   32                   // ceil(2001/64)
#define NSEG        6

// wrap-increment mod 3 (branchless -> v_cndmask)
__device__ __forceinline__ int inc3(int t) { return (t == 2) ? 0 : t + 1; }

// Branchless segment id. r = sequence row, t = r % 3.
// left rows [0,1000): seg = t ; right rows [1001,2001): seg = 3 + (t+1)%3 ;
// row 1000 (overlap) and rows >= 2001 (tail pad): seg = 99 (matches no lane).
__device__ __forceinline__ int seg_from(int r, int t) {
    int seg = (r < 1000) ? t : (3 + inc3(t));
    seg = (r == 1000) ? 99 : seg;
    seg = (r >= 2001) ? 99 : seg;
    return seg;
}

__device__ __forceinline__ float onehot(int seg, int m) {
    return (seg == m) ? 1.0f : 0.0f;
}

__global__ __launch_bounds__(256)
void seg_mean_wmma_kernel(const float* __restrict__ in, float* __restrict__ out) {
    __shared__ float tile[TILE_ELEMS];   // 64 rows x 16 models
    __shared__ float acc[NSEG * M_DIM];  // cross-wave segment sums

    const int tid  = threadIdx.x;
    const int lane = tid & 31;
    const int wave = tid >> 5;           // 0..7
    const int half = lane >> 4;          // which 16-lane half -> K block
    const int m    = lane & 15;          // A: row M ; B: col N
    const size_t seq = blockIdx.x;
    const float* __restrict__ src = in + seq * (size_t)SEQ_ELEMS;

    if (tid < NSEG * M_DIM) acc[tid] = 0.0f;

    // This lane's four row slots per tile: rA0+{0,1,4,5}; rA0 advances by 64.
    int rA0 = wave * 8 + 2 * half;       // sequence row of (chunk cc=0, VGPR slot x)
    int t   = rA0 % 3;                   // computed once; then wrap-incremented

    // Loop-invariant LDS word indices for this lane's B reads.
    const int l0 = rA0 * M_DIM + m;      // cc=0, k slot x
    // cc=0 slot y: +16 ; cc=1 slot x: +64 ; cc=1 slot y: +80 (immediate offsets)

    v8f c = {};  // 16x16 f32 accumulator; rows 0..5 = segment sums

    for (int it = 0; it < NUM_TILES; ++it) {
        // ---- stream 64 rows (4KB contiguous) into LDS, 16B/thread ----
        int eoff = it * TILE_ELEMS + tid * 4;
        int ld   = (eoff > SEQ_ELEMS - 4) ? (SEQ_ELEMS - 4) : eoff;  // clamp in-bounds
        v4f v = *(const v4f*)(src + ld);
        __syncthreads();                       // previous tile fully consumed
        *(v4f*)(tile + tid * 4) = v;
        __syncthreads();                       // tile ready

        // mod-3 values for rows rA0, rA0+1 (== rA0+4), rA0+5
        const int t1 = inc3(t);
        const int t2 = inc3(t1);

        // ---- chunk cc=0 : rows rA0, rA0+1 ----
        {
            v2f a, b;
            a.x = onehot(seg_from(rA0,     t ), m);
            a.y = onehot(seg_from(rA0 + 1, t1), m);
            b.x = tile[l0];
            b.y = tile[l0 + 16];
            c = __builtin_amdgcn_wmma_f32_16x16x4_f32(
                    false, a, false, b, (short)0, c, false, false);
        }
        // ---- chunk cc=1 : rows rA0+4, rA0+5 ----
        {
            v2f a, b;
            a.x = onehot(seg_from(rA0 + 4, t1), m);
            a.y = onehot(seg_from(rA0 + 5, t2), m);
            b.x = tile[l0 + 64];
            b.y = tile[l0 + 80];
            c = __builtin_amdgcn_wmma_f32_16x16x4_f32(
                    false, a, false, b, (short)0, c, false, false);
        }

        rA0 += ROWS_TILE;                 // +64
        t = inc3(t);                      // 64 % 3 == 1
    }

    // ---- combine 8 per-wave accumulators: segment s lives in VGPR s of
    //      lanes 0..15, column N = lane ----
    __syncthreads();
    if (lane < 16) {
        atomicAdd(&acc[0 * M_DIM + m], c[0]);
        atomicAdd(&acc[1 * M_DIM + m], c[1]);
        atomicAdd(&acc[2 * M_DIM + m], c[2]);
        atomicAdd(&acc[3 * M_DIM + m], c[3]);
        atomicAdd(&acc[4 * M_DIM + m], c[4]);
        atomicAdd(&acc[5 * M_DIM + m], c[5]);
    }
    __syncthreads();

    // ---- mean + store: counts are compile-time constants (never zero) ----
    if (tid < NSEG * M_DIM) {
        const int s = tid >> 4;
        const float cnt = (s == 0 || s == 3) ? 334.0f : 333.0f;
        out[seq * (NSEG * M_DIM) + tid] = acc[tid] / cnt;
    }
}

extern "C" void kernel_launch(void* const* d_in, const int* in_sizes, int n_in,
                              void* d_out, int out_size, void* d_ws, size_t ws_size,
                              hipStream_t stream) {
    const float* in  = (const float*)d_in[0];   // (2048*2001, 16) f32
    float*       out = (float*)d_out;           // (2048, 96) f32
    (void)in_sizes; (void)n_in; (void)out_size; (void)d_ws; (void)ws_size;
    seg_mean_wmma_kernel<<<dim3(2048), dim3(256), 0, stream>>>(in, out);
}